// RegistrationLoss_12000138625091
// MI455X (gfx1250) — compile-verified
//
#include <hip/hip_runtime.h>

typedef float v2f __attribute__((ext_vector_type(2)));
typedef float v8f __attribute__((ext_vector_type(8)));

#define Bn 2
#define Dn 160
#define Hn 160
#define Wn 160
#define SLICE (Hn * Wn)            // 25600
#define VOL   (Dn * SLICE)         // 4,096,000
#define NFIELD 5
#define PAD 4
#define WINSZ 729.0f               // 9^3

// workspace layout (in floats)
#define FIELDS_SZ ((size_t)NFIELD * Bn * VOL)  // 40,960,000 floats = 163.84 MB
#define CC_SLOTS  (Bn * Hn)                    // 320
#define GRAD_WGS  1024
#define GP_SLOTS  (GRAD_WGS * 3)

__device__ __forceinline__ float band01(int r, int kr) {
  int d = r - kr;
  if (d < 0) d = -d;
  return (d <= PAD) ? 1.0f : 0.0f;
}

// CDNA5 async Global->LDS copy (ASYNCcnt-tracked, no VGPR round trip).
// %0 = LDS byte address (VGPR), %1 = 64-bit global address (VGPR pair).
__device__ __forceinline__ void async_copy_b128(unsigned lds_byte, const float* g) {
  asm volatile("global_load_async_to_lds_b128 %0, %1, off"
               :: "v"(lds_byte), "v"(g)
               : "memory");
}
__device__ __forceinline__ void wait_async0() {
  asm volatile("s_wait_asynccnt 0" ::: "memory");
}

// ---------------------------------------------------------------------------
// Stage A: per (b,z) slice — W filter in LDS, H filter as banded-ones WMMA
// fields[f][b][z][h][w], f in {p, t, p^2, t^2, p*t} (each W,H box-filtered)
// ---------------------------------------------------------------------------
__global__ __launch_bounds__(512)
void ncc_wh_kernel(const float* __restrict__ moved,
                   const float* __restrict__ fixedp,
                   float* __restrict__ fields) {
  extern __shared__ float smem[];
  float* sM = smem;               // 25600 floats
  float* sF = smem + SLICE;       // 25600
  float* sT = smem + 2 * SLICE;   // 25600 (W-filtered field)

  const int wg = blockIdx.x;      // 0..Bn*Dn-1
  const int b  = wg / Dn;
  const int z  = wg % Dn;
  const float* mp = moved  + (size_t)b * VOL + (size_t)z * SLICE;
  const float* fp = fixedp + (size_t)b * VOL + (size_t)z * SLICE;

  const int tid = threadIdx.x;

  // ---- async-stage both input slices into LDS (B128 granularity) ----
  const unsigned ldsM = (unsigned)(uintptr_t)sM;
  const unsigned ldsF = (unsigned)(uintptr_t)sF;
  for (int i = tid; i < SLICE / 4; i += 512) {   // 6400 float4 per buffer
    async_copy_b128(ldsM + 16u * (unsigned)i, mp + 4 * i);
    async_copy_b128(ldsF + 16u * (unsigned)i, fp + 4 * i);
  }
  wait_async0();
  __syncthreads();

  const int lane = tid & 31;
  const int half = lane >> 4;     // 0 | 1
  const int l16  = lane & 15;
  // wave id as an SGPR so tile/K loops are scalar and WMMA runs full-EXEC
  const int wave = __builtin_amdgcn_readfirstlane(tid >> 5);  // 0..15

  for (int f = 0; f < NFIELD; ++f) {
    // ---- W-direction 9-tap box filter (zero padded) into sT ----
    for (int i = tid; i < SLICE; i += 512) {
      const int h = i / Wn;
      const int w = i - h * Wn;
      const int base = h * Wn;
      int wlo = w - PAD; if (wlo < 0) wlo = 0;
      int whi = w + PAD; if (whi > Wn - 1) whi = Wn - 1;
      float s = 0.0f;
      for (int ww = wlo; ww <= whi; ++ww) {
        const float m = sM[base + ww];
        const float x = sF[base + ww];
        float v;
        switch (f) {
          case 0:  v = m;     break;
          case 1:  v = x;     break;
          case 2:  v = m * m; break;
          case 3:  v = x * x; break;
          default: v = m * x; break;
        }
        s += v;
      }
      sT[i] = s;
    }
    __syncthreads();

    // ---- H-direction filter: Out = Band(160x160, ones width 9) * sT ----
    // 16x16 output tiles, K in blocks of 4; only |h-k|<=4 band blocks touched.
    float* out = fields + ((size_t)(f * Bn + b) * Dn + z) * SLICE;
    for (int t = wave; t < 100; t += 16) {        // scalar loop, EXEC all-ones
      const int I = t / 10, J = t % 10;
      const int h0 = I * 16, w0 = J * 16;
      v8f c = {};
      int kb0 = 4 * I - 1; if (kb0 < 0)  kb0 = 0;
      int kb1 = 4 * I + 4; if (kb1 > 39) kb1 = 39;
      for (int kb = kb0; kb <= kb1; ++kb) {
        const int kA  = half * 2;            // this lane holds K = kA, kA+1
        const int r   = h0 + l16;            // A-matrix row (M)
        const int kr0 = kb * 4 + kA;
        const int kr1 = kr0 + 1;
        v2f a, bm;
        a.x  = band01(r, kr0);
        a.y  = band01(r, kr1);
        bm.x = sT[kr0 * Wn + w0 + l16];      // B[k][n], n striped across lanes
        bm.y = sT[kr1 * Wn + w0 + l16];
        c = __builtin_amdgcn_wmma_f32_16x16x4_f32(
            /*neg_a=*/false, a, /*neg_b=*/false, bm,
            /*c_mod=*/(short)0, c, /*reuse_a=*/false, /*reuse_b=*/false);
      }
      // C/D layout: VGPR r -> M = h0 + r (+8 for upper lane half), N across lanes
      const int n     = w0 + l16;
      const int mbase = h0 + half * 8;
#pragma unroll
      for (int r = 0; r < 8; ++r) out[(mbase + r) * Wn + n] = c[r];
    }
    __syncthreads();
  }
}

// ---------------------------------------------------------------------------
// Stage B: D-direction running-window sums + pointwise NCC + block reduction
// ---------------------------------------------------------------------------
__global__ __launch_bounds__(160)
void ncc_d_reduce_kernel(const float* __restrict__ fields,
                         float* __restrict__ ccpart) {
  const int wg = blockIdx.x;            // b*Hn + h
  const int b  = wg / Hn;
  const int h  = wg % Hn;
  const int w  = threadIdx.x;           // 0..159

  const float* fb[NFIELD];
  for (int f = 0; f < NFIELD; ++f)
    fb[f] = fields + (size_t)(f * Bn + b) * VOL + (size_t)h * Wn + w;

  float S[NFIELD] = {0.f, 0.f, 0.f, 0.f, 0.f};
  for (int zz = 0; zz <= PAD; ++zz)
    for (int f = 0; f < NFIELD; ++f)
      S[f] += fb[f][(size_t)zz * SLICE];

  float acc = 0.0f;
  for (int z = 0; z < Dn; ++z) {
    if (z > 0) {
      const int zi = z + PAD;       // enters window
      const int zo = z - PAD - 1;   // leaves window (re-read: L2 resident)
      for (int f = 0; f < NFIELD; ++f) {
        const float add = (zi < Dn) ? fb[f][(size_t)zi * SLICE] : 0.0f;
        const float sub = (zo >= 0) ? fb[f][(size_t)zo * SLICE] : 0.0f;
        S[f] += add - sub;
      }
    }
    const float ps = S[0], ts = S[1], pss = S[2], tss = S[3], pts = S[4];
    const float pm = ps / WINSZ;
    const float tm = ts / WINSZ;
    const float cross = pts - tm * ps - pm * ts + pm * tm * WINSZ;
    float pv = pss - 2.0f * pm * ps + pm * pm * WINSZ; if (pv < 0.0f) pv = 0.0f;
    float tv = tss - 2.0f * tm * ts + tm * tm * WINSZ; if (tv < 0.0f) tv = 0.0f;
    float cc = (cross * cross) / (pv * tv + 0.001f);
    cc = cc < 0.0f ? 0.0f : (cc > 1.0f ? 1.0f : cc);
    acc += cc;
  }

  __shared__ float red[160];
  red[w] = acc;
  __syncthreads();
  for (int s = 80; s >= 5; s >>= 1) {          // 160->80->40->20->10->5
    if (w < s) red[w] += red[w + s];
    __syncthreads();
  }
  if (w == 0)
    ccpart[wg] = red[0] + red[1] + red[2] + red[3] + red[4];
}

// ---------------------------------------------------------------------------
// Stage C: flow gradient-loss partial sums
// ---------------------------------------------------------------------------
__global__ __launch_bounds__(256)
void grad_kernel(const float* __restrict__ flow, float* __restrict__ gpart) {
  const unsigned NT = (unsigned)Bn * 3u * (unsigned)VOL;   // 24,576,000
  float sy = 0.f, sx = 0.f, sz = 0.f;
  for (unsigned i = blockIdx.x * 256u + threadIdx.x; i < NT;
       i += (unsigned)GRAD_WGS * 256u) {
    const unsigned w  = i % Wn;
    const unsigned r1 = i / Wn;
    const unsigned h  = r1 % Hn;
    const unsigned r2 = r1 / Hn;
    const unsigned z  = r2 % Dn;
    const float v = flow[i];
    if (z < Dn - 1) sy += fabsf(flow[i + SLICE] - v);  // diff along D
    if (h < Hn - 1) sx += fabsf(flow[i + Wn] - v);     // diff along H
    if (w < Wn - 1) sz += fabsf(flow[i + 1] - v);      // diff along W
  }
  __shared__ float ry[256], rx[256], rz[256];
  const int t = threadIdx.x;
  ry[t] = sy; rx[t] = sx; rz[t] = sz;
  __syncthreads();
  for (int s = 128; s > 0; s >>= 1) {
    if (t < s) { ry[t] += ry[t + s]; rx[t] += rx[t + s]; rz[t] += rz[t + s]; }
    __syncthreads();
  }
  if (t == 0) {
    gpart[blockIdx.x * 3 + 0] = ry[0];
    gpart[blockIdx.x * 3 + 1] = rx[0];
    gpart[blockIdx.x * 3 + 2] = rz[0];
  }
}

// ---------------------------------------------------------------------------
// Stage D: deterministic final combine
// ---------------------------------------------------------------------------
__global__ __launch_bounds__(256)
void finalize_kernel(const float* __restrict__ ccpart,
                     const float* __restrict__ gpart,
                     float* __restrict__ out) {
  const int t = threadIdx.x;
  float cc = 0.f, gy = 0.f, gx = 0.f, gz = 0.f;
  for (int i = t; i < CC_SLOTS; i += 256) cc += ccpart[i];
  for (int i = t; i < GRAD_WGS; i += 256) {
    gy += gpart[3 * i + 0];
    gx += gpart[3 * i + 1];
    gz += gpart[3 * i + 2];
  }
  __shared__ float rc[256], ry[256], rx[256], rz[256];
  rc[t] = cc; ry[t] = gy; rx[t] = gx; rz[t] = gz;
  __syncthreads();
  for (int s = 128; s > 0; s >>= 1) {
    if (t < s) {
      rc[t] += rc[t + s]; ry[t] += ry[t + s];
      rx[t] += rx[t + s]; rz[t] += rz[t + s];
    }
    __syncthreads();
  }
  if (t == 0) {
    const float nvox   = (float)Bn * (float)VOL;                    // 8,192,000
    const float gdenom = (float)Bn * 3.0f * 159.0f * 160.0f * 160.0f;
    const float sim = 1.0f - rc[0] / nvox;
    const float reg = (ry[0] / gdenom + rx[0] / gdenom + rz[0] / gdenom) / 3.0f;
    out[0] = sim + reg;
  }
}

// ---------------------------------------------------------------------------
extern "C" void kernel_launch(void* const* d_in, const int* in_sizes, int n_in,
                              void* d_out, int out_size, void* d_ws, size_t ws_size,
                              hipStream_t stream) {
  const float* moved  = (const float*)d_in[0];
  const float* fixedp = (const float*)d_in[1];
  const float* flow   = (const float*)d_in[2];
  float* ws     = (float*)d_ws;
  float* fields = ws;                        // FIELDS_SZ floats
  float* ccpart = ws + FIELDS_SZ;            // CC_SLOTS floats
  float* gpart  = ws + FIELDS_SZ + CC_SLOTS; // GP_SLOTS floats
  float* out    = (float*)d_out;

  ncc_wh_kernel<<<Bn * Dn, 512, 3 * SLICE * sizeof(float), stream>>>(
      moved, fixedp, fields);
  ncc_d_reduce_kernel<<<Bn * Hn, 160, 0, stream>>>(fields, ccpart);
  grad_kernel<<<GRAD_WGS, 256, 0, stream>>>(flow, gpart);
  finalize_kernel<<<1, 256, 0, stream>>>(ccpart, gpart, out);
}